// PaiIndexMatrix_63763084476460
// MI455X (gfx1250) — compile-verified
//
#include <hip/hip_runtime.h>

typedef __attribute__((ext_vector_type(2))) float v2f;
typedef __attribute__((ext_vector_type(8))) float v8f;

#define NPTS  4096
#define BATCH 8
#define KNN   20
#define KSZ   32

// ---------------------------------------------------------------------------
// Kernel 0: squared norms of every point -> workspace (B*N floats = 128 KB)
// ---------------------------------------------------------------------------
__global__ __launch_bounds__(256) void norms_kernel(const float* __restrict__ x,
                                                    float* __restrict__ norms) {
    int i = blockIdx.x * blockDim.x + threadIdx.x;
    if (i >= BATCH * NPTS) return;
    int b = i >> 12;              // N = 4096
    int n = i & (NPTS - 1);
    const float* xb = x + b * 3 * NPTS;
    float a = xb[n], c = xb[NPTS + n], d = xb[2 * NPTS + n];
    norms[i] = a * a + c * c + d * d;
}

// ---------------------------------------------------------------------------
// Kernel 1: fused pairwise-distance (WMMA f32 16x16x4) + streaming top-20.
// One wave handles 16 query points; loops over candidate tiles, 2 tiles (32
// candidates) per iteration -> two independent v_wmma per loop body.
//   A (16x4)  = candidate tile coords (K=3 padded to 4)
//   B (4x16)  = query tile coords (wave-invariant)
//   D (16x16) = inner products; lane owns query n=lane&15, 8 candidate rows.
// Operand loads are branch-free: per-lane base pointers + a 0/1 multiplier
// replace the hi/lo ternary so no EXEC churn in the hot loop.
// ---------------------------------------------------------------------------
__global__ __launch_bounds__(128) void knn_topk_kernel(const float* __restrict__ x,
                                                       const float* __restrict__ norms,
                                                       int* __restrict__ out_idx) {
    const int lane  = threadIdx.x & 31;
    const int wave  = blockIdx.x * (blockDim.x >> 5) + (threadIdx.x >> 5);
    const int b     = wave >> 8;                 // 256 waves per batch
    const int q0    = (wave & 255) << 4;         // 16 queries per wave
    const float* xb = x + b * 3 * NPTS;
    const float* nb = norms + b * NPTS;

    const int   li    = lane & 15;
    const bool  hi    = lane >= 16;
    const int   hioff = hi ? 8 : 0;
    const float sely  = hi ? 0.0f : 1.0f;        // zero out K=3 pad lane half

    // per-lane operand base pointers: lane<16 reads dims {0,1}, lane>=16 {2,pad}
    const float* apx = xb + (hi ? 2 * NPTS : 0) + li;
    const float* apy = xb + NPTS + li;

    // B operand (queries), wave-invariant
    v2f bmat;
    bmat.x = apx[q0];
    bmat.y = apy[q0] * sely;
    const float qq = nb[q0 + li];                // ||query||^2 for n = lane&15

    float vals[KNN];
    int   idxs[KNN];
#pragma unroll
    for (int j = 0; j < KNN; ++j) { vals[j] = -3.0e38f; idxs[j] = 0; }

    for (int c0 = 0; c0 < NPTS; c0 += 32) {
        // two candidate tiles, branch-free loads
        v2f a0, a1;
        a0.x = apx[c0];      a0.y = apy[c0] * sely;
        a1.x = apx[c0 + 16]; a1.y = apy[c0 + 16] * sely;

        v8f acc0 = {}, acc1 = {};
        acc0 = __builtin_amdgcn_wmma_f32_16x16x4_f32(
            false, a0, false, bmat, (short)0, acc0, false, false);
        acc1 = __builtin_amdgcn_wmma_f32_16x16x4_f32(
            false, a1, false, bmat, (short)0, acc1, false, false);

        // candidate norms for this lane's 2x8 rows (contiguous float4s)
        const float4 cn0 = *(const float4*)(nb + c0 + hioff);
        const float4 cn1 = *(const float4*)(nb + c0 + hioff + 4);
        const float4 cn2 = *(const float4*)(nb + c0 + 16 + hioff);
        const float4 cn3 = *(const float4*)(nb + c0 + 16 + hioff + 4);
        float cns[16] = {cn0.x, cn0.y, cn0.z, cn0.w, cn1.x, cn1.y, cn1.z, cn1.w,
                         cn2.x, cn2.y, cn2.z, cn2.w, cn3.x, cn3.y, cn3.z, cn3.w};

#pragma unroll
        for (int v = 0; v < 16; ++v) {
            // pd = 2<q,c> - ||q||^2 - ||c||^2 ; larger = closer
            float inner = (v < 8) ? acc0[v & 7] : acc1[v & 7];
            float d     = 2.0f * inner - qq - cns[v];
            int   ci    = c0 + hioff + (v & 7) + ((v < 8) ? 0 : 16);
            if (d > vals[KNN - 1]) {             // guarded unrolled insertion
                float cv = d; int cidx = ci;
#pragma unroll
                for (int j = 0; j < KNN; ++j) {
                    bool  gt = cv > vals[j];
                    float tv = vals[j]; int ti = idxs[j];
                    vals[j] = gt ? cv : vals[j];
                    idxs[j] = gt ? cidx : idxs[j];
                    cv = gt ? tv : cv;  cidx = gt ? ti : cidx;
                }
            }
        }
    }

    // merge lane-pair lists: snapshot partner's 20 (static-index shuffles),
    // then insert into own sorted list.
    float ovals[KNN]; int oidxs[KNN];
#pragma unroll
    for (int j = 0; j < KNN; ++j) {
        ovals[j] = __shfl_xor(vals[j], 16, 32);
        oidxs[j] = __shfl_xor(idxs[j], 16, 32);
    }
#pragma unroll
    for (int j = 0; j < KNN; ++j) {
        if (ovals[j] > vals[KNN - 1]) {
            float cv = ovals[j]; int cidx = oidxs[j];
#pragma unroll
            for (int t = 0; t < KNN; ++t) {
                bool  gt = cv > vals[t];
                float tv = vals[t]; int ti = idxs[t];
                vals[t] = gt ? cv : vals[t];
                idxs[t] = gt ? cidx : idxs[t];
                cv = gt ? tv : cv;  cidx = gt ? ti : cidx;
            }
        }
    }

    if (lane < 16) {
        int* o = out_idx + (size_t)(b * NPTS + q0 + lane) * KNN;
#pragma unroll
        for (int j = 0; j < KNN; ++j) o[j] = b * NPTS + idxs[j];
    }
}

// ---------------------------------------------------------------------------
// Kernel 2: per-point [20,3]x[3,32] matmul + top_max normalization.
// One wave per point; lane = KS column (exactly 32). Normalization axis (k=20)
// is the lane-local loop axis -> no cross-lane reductions. Neighbor coords are
// loaded once by lanes 0..19 and broadcast with static shuffles.
// ---------------------------------------------------------------------------
__global__ __launch_bounds__(128) void adjweight_kernel(const float* __restrict__ x,
                                                        const float* __restrict__ kernals,
                                                        const float* __restrict__ one_padding,
                                                        const int* __restrict__ sp_idx,
                                                        float* __restrict__ out_w) {
    const int lane = threadIdx.x & 31;
    const int g    = blockIdx.x * (blockDim.x >> 5) + (threadIdx.x >> 5); // point id
    const int b    = g >> 12;
    const float* xb = x + b * 3 * NPTS;

    int myj = lane < KNN ? lane : (KNN - 1);
    int nbr = sp_idx[(size_t)g * KNN + myj];
    int n   = nbr & (NPTS - 1);
    float px = xb[n], py = xb[NPTS + n], pz = xb[2 * NPTS + n];

    float p0x = __shfl(px, 0, 32), p0y = __shfl(py, 0, 32), p0z = __shfl(pz, 0, 32);
    float kx = kernals[lane], ky = kernals[KSZ + lane], kz = kernals[2 * KSZ + lane];

    float w[KNN];
    float s1 = 0.0f;
#pragma unroll
    for (int j = 0; j < KNN; ++j) {
        float dx = __shfl(px, j, 32) - p0x;
        float dy = __shfl(py, j, 32) - p0y;
        float dz = __shfl(pz, j, 32) - p0z;
        float t  = dx * kx + dy * ky + dz * kz + one_padding[j * KSZ + lane];
        t = t > 0.0f ? t : 0.0f;                  // relu
        w[j] = t; s1 += t;
    }
    float inv1 = 1.0f / (s1 + 1e-6f);
    float s2 = 0.0f;
#pragma unroll
    for (int j = 0; j < KNN; ++j) { float t = w[j] * inv1; t = t * t; w[j] = t; s2 += t; }
    float inv2 = 1.0f / (s2 + 1e-6f);

    float* ow = out_w + (size_t)g * (KNN * KSZ);
#pragma unroll
    for (int j = 0; j < KNN; ++j) {
        float t = w[j] * inv2;
        ow[j * KSZ + lane] = t > 0.1f ? t : 0.0f;
    }
}

// ---------------------------------------------------------------------------
extern "C" void kernel_launch(void* const* d_in, const int* in_sizes, int n_in,
                              void* d_out, int out_size, void* d_ws, size_t ws_size,
                              hipStream_t stream) {
    const float* x    = (const float*)d_in[0];   // [B,3,N]
    const float* kern = (const float*)d_in[1];   // [3,32]
    const float* pad  = (const float*)d_in[2];   // [20,32]
    // d_in[3] = k (always 20, baked in)

    float* norms   = (float*)d_ws;                               // B*N floats
    int*   out_idx = (int*)d_out;                                // B*N*K int32
    float* out_w   = (float*)d_out + (size_t)BATCH * NPTS * KNN; // B*N*K*KS f32

    norms_kernel<<<(BATCH * NPTS + 255) / 256, 256, 0, stream>>>(x, norms);

    // 2048 waves total (16 queries each), 4 waves/block -> 512 blocks
    knn_topk_kernel<<<(BATCH * NPTS / 16) / 4, 128, 0, stream>>>(x, norms, out_idx);

    // one wave per point, 4 waves/block -> 8192 blocks
    adjweight_kernel<<<(BATCH * NPTS) / 4, 128, 0, stream>>>(x, kern, pad, out_idx, out_w);
}